// MoE_lora_88424786690148
// MI455X (gfx1250) — compile-verified
//
#include <hip/hip_runtime.h>
#include <hip/hip_bf16.h>

// ---------------------------------------------------------------------------
// MoE conv (8 experts top-2 + shared) for MI455X / gfx1250, wave32, WMMA f32.
//
// Sizes: B=32, C1=3, H=W=224, C2=64, E=8, K=7, stride=4, pad=3 -> OH=OW=56.
// Conv as implicit GEMM: M=64 out-channels, K=147 (padded 148), N-blocked:
// 64 pixels per workgroup = 4 x 16-pixel WMMA sub-tiles per wave, so every
// A (weight) fragment feeds 4 independent v_wmma_f32_16x16x4_f32 chains.
// fp32 end-to-end: the problem is HBM-bound (~50MB vs 5.7 GFLOP), so no
// precision downgrade is needed to reach the bandwidth roof.
// ---------------------------------------------------------------------------

typedef __attribute__((ext_vector_type(2))) float v2f;
typedef __attribute__((ext_vector_type(8))) float v8f;

#define BATCH   32
#define C1      3
#define HW_IN   224
#define C2      64
#define NEXP    8
#define KS      7
#define OHW     56
#define NPIX    (OHW * OHW)      // 3136
#define KDIM    147              // C1*KS*KS
#define KP      148              // padded to multiple of 4
#define NB      4                // N sub-tiles per workgroup (64 pixels)
#define NPIXWG  (NB * 16)        // 64
#define NTILE   (NPIX / NPIXWG)  // 49
#define LSTR    65               // LDS row stride for [pixel][channel] buffer
#define LN_EPS  1e-6f
#define LOSS_COEF 1e-2f

// ---------------------------------------------------------------------------
// Kernel 1: per-(batch,channel) mean over 224*224 pixels. grid=96, block=256.
// ---------------------------------------------------------------------------
__global__ __launch_bounds__(256) void k_chanmean(const float* __restrict__ x,
                                                  float* __restrict__ means) {
    const int bc = blockIdx.x;                 // b*3 + c
    const float* p = x + (size_t)bc * (HW_IN * HW_IN);
    float s = 0.f;
    for (int i = threadIdx.x; i < HW_IN * HW_IN; i += 256) s += p[i];
    __shared__ float red[256];
    red[threadIdx.x] = s;
    __syncthreads();
    for (int off = 128; off > 0; off >>= 1) {
        if (threadIdx.x < off) red[threadIdx.x] += red[threadIdx.x + off];
        __syncthreads();
    }
    if (threadIdx.x == 0) means[bc] = red[0] * (1.0f / (HW_IN * HW_IN));
}

// ---------------------------------------------------------------------------
// Kernel 2: gating + aux loss. 1 block of 32 threads (one wave, lane = batch).
// ---------------------------------------------------------------------------
__global__ __launch_bounds__(32) void k_gate(const float* __restrict__ means,
                                             const float* __restrict__ wg,   // [3][8]
                                             int*   __restrict__ top_idx,    // [32][2]
                                             float* __restrict__ top_g,      // [32][2]
                                             float* __restrict__ loss_out) {
    __shared__ float gates_s[BATCH][NEXP];
    __shared__ float impS[NEXP], loadS[NEXP];
    const int b = threadIdx.x;

    const float m0 = means[b * 3 + 0];
    const float m1 = means[b * 3 + 1];
    const float m2 = means[b * 3 + 2];
    float l[NEXP];
#pragma unroll
    for (int e = 0; e < NEXP; ++e)
        l[e] = m0 * wg[0 * NEXP + e] + m1 * wg[1 * NEXP + e] + m2 * wg[2 * NEXP + e];

    // top-2 (largest first; ties -> lower index, matching lax.top_k)
    int i0 = 0;
#pragma unroll
    for (int e = 1; e < NEXP; ++e) if (l[e] > l[i0]) i0 = e;
    int i1 = (i0 == 0) ? 1 : 0;
#pragma unroll
    for (int e = 0; e < NEXP; ++e) if (e != i0 && l[e] > l[i1]) i1 = e;

    // softmax over the two kept logits (l[i0] is the max)
    const float ex = __expf(l[i1] - l[i0]);
    const float g0 = 1.0f / (1.0f + ex);
    const float g1 = ex / (1.0f + ex);

#pragma unroll
    for (int e = 0; e < NEXP; ++e) gates_s[b][e] = 0.f;
    gates_s[b][i0] = g0;
    gates_s[b][i1] = g1;
    top_idx[2 * b + 0] = i0;
    top_idx[2 * b + 1] = i1;
    top_g[2 * b + 0] = g0;
    top_g[2 * b + 1] = g1;
    __syncthreads();

    if (b < NEXP) {
        float imp = 0.f, cnt = 0.f;
        for (int j = 0; j < BATCH; ++j) {
            const float g = gates_s[j][b];
            imp += g;
            if (g > 0.f) cnt += 1.f;
        }
        impS[b] = imp;
        loadS[b] = cnt;
    }
    __syncthreads();

    if (b == 0) {
        float loss = 0.f;
#pragma unroll
        for (int which = 0; which < 2; ++which) {
            const float* v = which ? loadS : impS;
            float mean = 0.f;
            for (int e = 0; e < NEXP; ++e) mean += v[e];
            mean *= (1.0f / NEXP);
            float var = 0.f;
            for (int e = 0; e < NEXP; ++e) {
                const float d = v[e] - mean;
                var += d * d;
            }
            var *= (1.0f / (NEXP - 1));          // ddof=1
            loss += var / (mean * mean + 1e-10f);
        }
        *loss_out = LOSS_COEF * loss;
    }
}

// ---------------------------------------------------------------------------
// Kernel 3: conv + LayerNorm + gated combine.
// grid = (49 tiles, 32 batches), block = 128 threads (4 waves).
// Wave w handles out channels 16w..16w+15 across NB=4 16-pixel sub-tiles,
// for all 3 experts (shared + top-2). Each A fragment feeds 4 WMMAs.
//
// WMMA f32 16x16x4 per-lane layouts (codegen-verified shape, ISA 7.12.2):
//   A (16x4, MxK): lane (half = lane>>4, mr = lane&15) holds row mr:
//                  vreg0 = A[mr][k0+2*half], vreg1 = A[mr][k0+1+2*half]
//   B (4x16, KxN): lane holds column mr:
//                  vreg0 = B[k0+2*half][mr], vreg1 = B[k0+1+2*half][mr]
//   C/D (16x16):   vreg j = D[j + 8*half][mr]
// ---------------------------------------------------------------------------
__global__ __launch_bounds__(128) void k_conv(
    const float* __restrict__ x,
    const float* __restrict__ eW,  const float* __restrict__ eb,
    const float* __restrict__ elw, const float* __restrict__ elb,
    const float* __restrict__ sW,  const float* __restrict__ sb,
    const float* __restrict__ slw, const float* __restrict__ slb,
    const int*   __restrict__ top_idx, const float* __restrict__ top_g,
    float* __restrict__ out) {

    __shared__ float patch[NPIXWG * KP];   // [pixel][k], K zero-padded: 37,888 B
    __shared__ float expbuf[NPIXWG * LSTR];// [pixel][channel] staging: 16,640 B

    const int t   = blockIdx.x;            // 64-pixel tile 0..48
    const int b   = blockIdx.y;            // batch
    const int tid = threadIdx.x;

    // ---- build im2col patch once (shared across all 3 experts) ----
    for (int i = tid; i < NPIXWG * KP; i += 128) {
        const int n = i / KP;
        const int k = i - n * KP;
        float v = 0.f;
        if (k < KDIM) {
            const int c  = k / (KS * KS);
            const int r  = k - c * (KS * KS);
            const int kh = r / KS;
            const int kw = r - kh * KS;
            const int p  = t * NPIXWG + n;
            const int oh = p / OHW;
            const int ow = p - oh * OHW;
            const int ih = oh * 4 - 3 + kh;
            const int iw = ow * 4 - 3 + kw;
            if (ih >= 0 && ih < HW_IN && iw >= 0 && iw < HW_IN)
                v = x[(((size_t)b * C1 + c) * HW_IN + ih) * HW_IN + iw];
        }
        patch[i] = v;
    }
    __syncthreads();

    const int   e0 = top_idx[2 * b + 0];
    const int   e1 = top_idx[2 * b + 1];
    const float g0 = top_g[2 * b + 0];
    const float g1 = top_g[2 * b + 1];

    const int wave  = tid >> 5;            // 0..3 -> M tile
    const int lane  = tid & 31;
    const int half  = lane >> 4;           // 0/1
    const int mr    = lane & 15;           // row (A) / column (B/C/D)
    const int mbase = wave * 16;

    // This thread's fixed LayerNorm/combine slice: pixel group + channel octet
    const int pxg = tid >> 3;              // 0..15 (pixel within 16-px group)
    const int sub = tid & 7;               // 0..7 -> channels sub*8..sub*8+7
    float accr[NB][8];                     // gated output accumulator (registers)
#pragma unroll
    for (int s = 0; s < NB; ++s)
#pragma unroll
        for (int j = 0; j < 8; ++j) accr[s][j] = 0.f;

    for (int exi = 0; exi < 3; ++exi) {
        const float *W, *bias, *lw, *lb;
        float coef;
        if (exi == 0) { W = sW; bias = sb; lw = slw; lb = slb; coef = 1.f; }
        else {
            const int e = (exi == 1) ? e0 : e1;
            W    = eW  + (size_t)e * C2 * KDIM;
            bias = eb  + e * C2;
            lw   = elw + e * C2;
            lb   = elb + e * C2;
            coef = (exi == 1) ? g0 : g1;
        }
        const float* Wrow = W + (size_t)(mbase + mr) * KDIM;

        v8f c[NB];
#pragma unroll
        for (int s = 0; s < NB; ++s) c[s] = (v8f){};

        // full chunks: k0 = 0,4,...,140 — one A fragment drives NB WMMAs
        for (int kc = 0; kc < 36; ++kc) {
            const int k0 = kc * 4 + 2 * half;
            v2f a;
            a.x = Wrow[k0];
            a.y = Wrow[k0 + 1];
#pragma unroll
            for (int s = 0; s < NB; ++s) {
                v2f bv;
                bv.x = patch[(s * 16 + mr) * KP + k0];
                bv.y = patch[(s * 16 + mr) * KP + k0 + 1];
                c[s] = __builtin_amdgcn_wmma_f32_16x16x4_f32(
                           false, a, false, bv, (short)0, c[s], false, false);
            }
        }
        // tail chunk k0=144..147 (weight index 147 past K=147 -> zero)
        {
            const int k0 = 144 + 2 * half;
            v2f a;
            a.x = Wrow[k0];
            a.y = (k0 + 1 < KDIM) ? Wrow[k0 + 1] : 0.f;
#pragma unroll
            for (int s = 0; s < NB; ++s) {
                v2f bv;
                bv.x = patch[(s * 16 + mr) * KP + k0];
                bv.y = patch[(s * 16 + mr) * KP + k0 + 1]; // zero-padded
                c[s] = __builtin_amdgcn_wmma_f32_16x16x4_f32(
                           false, a, false, bv, (short)0, c[s], false, false);
            }
        }

        // bias + stash this expert's 64ch x 64px block into [pixel][channel] LDS
#pragma unroll
        for (int s = 0; s < NB; ++s)
#pragma unroll
            for (int j = 0; j < 8; ++j) {
                const int ch = mbase + j + 8 * half;
                expbuf[(s * 16 + mr) * LSTR + ch] = c[s][j] + bias[ch];
            }
        __syncthreads();

        // LayerNorm over 64 channels per pixel: 8 lanes per pixel,
        // 4 pixel-blocks per thread. Gated accumulate into registers.
#pragma unroll
        for (int s = 0; s < NB; ++s) {
            const int px = s * 16 + pxg;
            float sum = 0.f, sq = 0.f;
#pragma unroll
            for (int j = 0; j < 8; ++j) {
                const float v = expbuf[px * LSTR + sub * 8 + j];
                sum += v;
                sq  += v * v;
            }
#pragma unroll
            for (int m = 1; m < 8; m <<= 1) {   // wave32: groups of 8 lanes
                sum += __shfl_xor(sum, m);
                sq  += __shfl_xor(sq, m);
            }
            const float u    = sum * (1.0f / C2);
            const float var  = sq * (1.0f / C2) - u * u;
            const float rstd = rsqrtf(var + LN_EPS);
#pragma unroll
            for (int j = 0; j < 8; ++j) {
                const int ch = sub * 8 + j;
                const float v = expbuf[px * LSTR + ch];
                accr[s][j] += coef * ((v - u) * rstd * lw[ch] + lb[ch]);
            }
        }
        __syncthreads();                   // expbuf reused by next expert
    }

    // ---- stage register accumulator through LDS for coalesced store ----
#pragma unroll
    for (int s = 0; s < NB; ++s) {
        const int px = s * 16 + pxg;
#pragma unroll
        for (int j = 0; j < 8; ++j)
            expbuf[px * LSTR + sub * 8 + j] = accr[s][j];
    }
    __syncthreads();

    // out[b][ch][t*64 + n]: 128 lanes sweep n fastest -> coalesced
    for (int i = tid; i < C2 * NPIXWG; i += 128) {
        const int ch = i >> 6;             // /64
        const int n  = i & 63;
        out[((size_t)b * C2 + ch) * NPIX + t * NPIXWG + n] = expbuf[n * LSTR + ch];
    }
}

// ---------------------------------------------------------------------------
// Launcher
// ---------------------------------------------------------------------------
extern "C" void kernel_launch(void* const* d_in, const int* in_sizes, int n_in,
                              void* d_out, int out_size, void* d_ws, size_t ws_size,
                              hipStream_t stream) {
    const float* x    = (const float*)d_in[0];
    const float* eW   = (const float*)d_in[1];
    const float* ebi  = (const float*)d_in[2];
    const float* elw  = (const float*)d_in[3];
    const float* elb  = (const float*)d_in[4];
    const float* sW   = (const float*)d_in[5];
    const float* sbi  = (const float*)d_in[6];
    const float* slw  = (const float*)d_in[7];
    const float* slb  = (const float*)d_in[8];
    const float* wg   = (const float*)d_in[9];

    float* out = (float*)d_out;

    // workspace: [0,384) channel means, [384,640) top idx, [640,896) top gates
    char*  ws      = (char*)d_ws;
    float* means   = (float*)(ws);
    int*   top_idx = (int*)(ws + 384);
    float* top_g   = (float*)(ws + 640);

    k_chanmean<<<dim3(BATCH * C1), dim3(256), 0, stream>>>(x, means);

    k_gate<<<dim3(1), dim3(32), 0, stream>>>(means, wg, top_idx, top_g,
                                             out + (out_size - 1));

    k_conv<<<dim3(NTILE, BATCH), dim3(128), 0, stream>>>(
        x, eW, ebi, elw, elb, sW, sbi, slw, slb, top_idx, top_g, out);
}